// GINEdgeAggregator_30288109371777
// MI455X (gfx1250) — compile-verified
//
#include <hip/hip_runtime.h>
#include <hip/hip_bf16.h>

typedef __attribute__((ext_vector_type(2))) float v2f;
typedef __attribute__((ext_vector_type(4))) float v4f;
typedef __attribute__((ext_vector_type(8))) float v8f;

#define BN_EPS_C 1e-5f

// ---------------------------------------------------------------------------
// Kernel 1: h0 = (1+eps)*x (float4 vectorized); zero BN stats accumulators.
// ---------------------------------------------------------------------------
__global__ void gin_init(const float* __restrict__ x, const float* __restrict__ eps,
                         float* __restrict__ h0, float* __restrict__ stats, int total4)
{
    int t = blockIdx.x * 256 + threadIdx.x;
    if (t < total4) {
        v4f xv = ((const v4f*)x)[t];
        ((v4f*)h0)[t] = xv * (1.0f + eps[0]);
    }
    if (blockIdx.x == 0 && threadIdx.x < 256) stats[threadIdx.x] = 0.0f;
}

// ---------------------------------------------------------------------------
// Kernel 2: edge scatter-add. One wave32 per edge, 2 channels per lane.
// x/agg are 25.6MB each -> resident in 192MB L2, so the random gathers and
// GLOBAL_ATOMIC_ADD_F32 resolve at L2 bandwidth, not HBM.
// ---------------------------------------------------------------------------
__global__ void gin_scatter(const float* __restrict__ x, const float* __restrict__ ew,
                            const long long* __restrict__ ei, float* __restrict__ h0, int E)
{
    long long t = (long long)blockIdx.x * 256 + threadIdx.x;
    int e = (int)(t >> 5);
    if (e >= E) return;
    int c = ((int)t & 31) * 2;
    long long s = ei[e];                 // src row
    long long d = ei[(long long)E + e];  // dst row
    float w = ew[e];
    v2f xv = *(const v2f*)(x + s * 64 + c);
    atomicAdd(&h0[d * 64 + c],     xv.x * w);
    atomicAdd(&h0[d * 64 + c + 1], xv.y * w);
}

// ---------------------------------------------------------------------------
// Kernel 3: out[N,64] = act(in)[N,64] @ W[64,64] + bias, fused per-channel
// sum/sumsq for BN. act = BN(scale,shift)+ReLU when APPLY_ACT!=0 (templated,
// branch-free inner loop). One wave computes a 16x64 stripe with
// V_WMMA_F32_16X16X4_F32 (f32-exact, memory-bound GEMM so no reason to drop
// precision). W staged in LDS pair-transposed so each B operand is a single
// aligned ds_load_b64 into a contiguous even VGPR pair (no repack movs):
//   sWt[(k/2)*128 + n*2 + (k&1)] = W[k][n]   ->  b = {W[k][n], W[k+1][n]}
// A layout (16x4 f32): lanes 0-15 -> K=k0,k0+1 ; lanes 16-31 -> K=k0+2,k0+3
// (one contiguous float2 per lane). C/D: v[i] = (M=i+8*kh, N=lane&15).
// ---------------------------------------------------------------------------
template <int APPLY_ACT>
__global__ void gin_gemm64(const float* __restrict__ in, const float* __restrict__ W,
                           const float* __restrict__ bias,
                           const float* __restrict__ in_scale, const float* __restrict__ in_shift,
                           float* __restrict__ out,
                           float* __restrict__ ssum, float* __restrict__ ssq,
                           int ntiles, int nrows)
{
    __shared__ float sWt[64 * 64];
    __shared__ float sSc[64];
    __shared__ float sSh[64];

    const int tid = threadIdx.x;
    for (int i = tid; i < 64 * 64; i += 256) {
        int k = i >> 6, n = i & 63;
        sWt[(k >> 1) * 128 + n * 2 + (k & 1)] = W[i];
    }
    if (APPLY_ACT && tid < 64) {
        sSc[tid] = in_scale[tid];
        sSh[tid] = in_shift[tid];
    }
    __syncthreads();

    const int wave = tid >> 5;
    const int lane = tid & 31;
    const int tile = blockIdx.x * 8 + wave;
    if (tile >= ntiles) return;

    const int nrow = lane & 15;   // M index (lanes 0-15) and N index for B/C/D
    const int kh   = lane >> 4;   // K-half select

    int arow_idx = tile * 16 + nrow;
    if (arow_idx >= nrows) arow_idx = nrows - 1;   // safe clamp (N%16==0 in practice)
    const float* arow = in + (long long)arow_idx * 64;

    v8f acc[4];
    #pragma unroll
    for (int nt = 0; nt < 4; ++nt) acc[nt] = (v8f){0,0,0,0,0,0,0,0};

    #pragma unroll
    for (int k0 = 0; k0 < 64; k0 += 4) {
        const int k = k0 + 2 * kh;
        v2f a = *(const v2f*)(arow + k);
        if (APPLY_ACT) {
            v2f sc = *(const v2f*)(sSc + k);
            v2f sh = *(const v2f*)(sSh + k);
            a.x = fmaxf(a.x * sc.x + sh.x, 0.0f);
            a.y = fmaxf(a.y * sc.y + sh.y, 0.0f);
        }
        const float* wp = sWt + (k >> 1) * 128 + nrow * 2;
        #pragma unroll
        for (int nt = 0; nt < 4; ++nt) {
            v2f b = *(const v2f*)(wp + nt * 32);
            acc[nt] = __builtin_amdgcn_wmma_f32_16x16x4_f32(
                false, a, false, b, (short)0, acc[nt], false, false);
        }
    }

    // Epilogue: bias, store, fused BN statistics. Cross-half shuffle reduction
    // halves the stat atomics (lanes L and L+16 share the same output column).
    #pragma unroll
    for (int nt = 0; nt < 4; ++nt) {
        const int col = nt * 16 + nrow;
        const float bcol = bias[col];
        float s = 0.0f, sq = 0.0f;
        #pragma unroll
        for (int i = 0; i < 8; ++i) {
            int m = tile * 16 + kh * 8 + i;
            float v = acc[nt][i] + bcol;
            if (m < nrows) {
                out[(long long)m * 64 + col] = v;
                s  += v;
                sq += v * v;
            }
        }
        s  += __shfl_xor(s, 16, 32);
        sq += __shfl_xor(sq, 16, 32);
        if (kh == 0) {
            atomicAdd(&ssum[col], s);
            atomicAdd(&ssq[col],  sq);
        }
    }
}

// ---------------------------------------------------------------------------
// Kernel 4: fold sum/sumsq into BN scale/shift (64 channels).
// ---------------------------------------------------------------------------
__global__ void gin_bnstats(const float* __restrict__ sum, const float* __restrict__ sq,
                            const float* __restrict__ gamma, const float* __restrict__ beta,
                            float* __restrict__ scale, float* __restrict__ shift, float invN)
{
    int c = threadIdx.x;
    if (c < 64) {
        float mean = sum[c] * invN;
        float var  = sq[c] * invN - mean * mean;
        float inv  = rsqrtf(var + BN_EPS_C);
        float sc   = gamma[c] * inv;
        scale[c] = sc;
        shift[c] = beta[c] - mean * sc;
    }
}

// ---------------------------------------------------------------------------
// Kernel 5: final BN+ReLU applied in place on d_out (float4 vectorized).
// ---------------------------------------------------------------------------
__global__ void gin_bnrelu(float* __restrict__ y, const float* __restrict__ scale,
                           const float* __restrict__ shift, int total4)
{
    int t = blockIdx.x * 256 + threadIdx.x;
    if (t >= total4) return;
    int cb = (t & 15) * 4;        // 64 channels = 16 float4 groups per row
    v4f v  = ((const v4f*)y)[t];
    v4f sc = *(const v4f*)(scale + cb);
    v4f sh = *(const v4f*)(shift + cb);
    v = v * sc + sh;
    v.x = fmaxf(v.x, 0.0f);
    v.y = fmaxf(v.y, 0.0f);
    v.z = fmaxf(v.z, 0.0f);
    v.w = fmaxf(v.w, 0.0f);
    ((v4f*)y)[t] = v;
}

// ---------------------------------------------------------------------------
extern "C" void kernel_launch(void* const* d_in, const int* in_sizes, int n_in,
                              void* d_out, int out_size, void* d_ws, size_t ws_size,
                              hipStream_t stream)
{
    const float*     x   = (const float*)d_in[0];
    const float*     ew  = (const float*)d_in[1];
    const float*     eps = (const float*)d_in[2];
    const float*     W1  = (const float*)d_in[3];
    const float*     b1  = (const float*)d_in[4];
    const float*     g1  = (const float*)d_in[5];
    const float*     be1 = (const float*)d_in[6];
    const float*     W2  = (const float*)d_in[7];
    const float*     b2  = (const float*)d_in[8];
    const float*     g2  = (const float*)d_in[9];
    const float*     be2 = (const float*)d_in[10];
    const long long* ei  = (const long long*)d_in[11];

    const int N = in_sizes[0] / 64;
    const int E = in_sizes[1];
    const int total  = N * 64;
    const int total4 = total >> 2;

    float* h0 = (float*)d_ws;                     // [N,64] aggregated features
    float* y1 = h0 + (long long)total;            // [N,64] layer-1 pre-BN output
    float* S  = y1 + (long long)total;            // stats block (512 floats)
    float* sum1 = S;        float* sq1 = S + 64;
    float* sum2 = S + 128;  float* sq2 = S + 192;
    float* sc1  = S + 256;  float* sh1 = S + 320;
    float* sc2  = S + 384;  float* sh2 = S + 448;
    float* yout = (float*)d_out;

    const int ntiles = (N + 15) / 16;             // 6250 (exact for N=100000)

    gin_init<<<(total4 + 255) / 256, 256, 0, stream>>>(x, eps, h0, S, total4);

    long long sthreads = (long long)E * 32;
    gin_scatter<<<(int)((sthreads + 255) / 256), 256, 0, stream>>>(x, ew, ei, h0, E);

    gin_gemm64<0><<<(ntiles + 7) / 8, 256, 0, stream>>>(h0, W1, b1, nullptr, nullptr,
                                                        y1, sum1, sq1, ntiles, N);
    gin_bnstats<<<1, 64, 0, stream>>>(sum1, sq1, g1, be1, sc1, sh1, 1.0f / (float)N);

    gin_gemm64<1><<<(ntiles + 7) / 8, 256, 0, stream>>>(y1, W2, b2, sc1, sh1,
                                                        yout, sum2, sq2, ntiles, N);
    gin_bnstats<<<1, 64, 0, stream>>>(sum2, sq2, g2, be2, sc2, sh2, 1.0f / (float)N);

    gin_bnrelu<<<(total4 + 255) / 256, 256, 0, stream>>>(yout, sc2, sh2, total4);
}